// ModelNew_67276367724700
// MI455X (gfx1250) — compile-verified
//
#include <hip/hip_runtime.h>
#include <stdint.h>

typedef __attribute__((ext_vector_type(2))) float v2f;
typedef __attribute__((ext_vector_type(8))) float v8f;

#define K_DIM 4096
#define N_DIM 4096
#define ROWS  8
#define LDS_STRIDE 4100                      // 4096 + 4 floats pad -> conflict-free banks
#define C_BASE 0                             // float offsets in smem
#define X_BASE K_DIM
#define P_BASE (X_BASE + ROWS * LDS_STRIDE)  // 8 waves * 8 rows partials
#define R_BASE (P_BASE + 64)
#define SMEM_FLOATS (R_BASE + ROWS)          // ~147.9 KB total -> 2 WGs / 320KB WGP

// ---- CDNA5 async global->LDS copy (ASYNCcnt path) ------------------------
__device__ __forceinline__ void async_ld_b128(uint32_t lds_byte_off, const float* g) {
  asm volatile("global_load_async_to_lds_b128 %0, %1, off"
               :: "v"(lds_byte_off), "v"(g) : "memory");
}
__device__ __forceinline__ void wait_async0() {
  asm volatile("s_wait_asynccnt 0x0" ::: "memory");
}

__device__ __forceinline__ float gelu_tanh(float v) {
  float inner = 0.7978845608028654f * (v + 0.044715f * v * v * v);
  return 0.5f * v * (1.0f + tanhf(inner));
}

// ---- kernel 1: bias_term = (sum(bias) - sum(subtract)) / N ---------------
__global__ void bias_term_kernel(const float* __restrict__ b,
                                 const float* __restrict__ s,
                                 float* __restrict__ ws, int n) {
  __shared__ float red[8];
  float acc = 0.f;
  for (int i = threadIdx.x; i < n; i += 256) acc += b[i] - s[i];
  for (int off = 16; off; off >>= 1) acc += __shfl_down(acc, off, 32);
  if ((threadIdx.x & 31) == 0) red[threadIdx.x >> 5] = acc;
  __syncthreads();
  if (threadIdx.x == 0) {
    float t = 0.f;
    for (int w = 0; w < 8; ++w) t += red[w];
    ws[0] = t * (1.0f / (float)N_DIM);
  }
}

// ---- kernel 2: c[k] = sum_n W[k,n]  (one wave per contiguous 16KB row) ---
__global__ void colsum_kernel(const float* __restrict__ W,
                              float* __restrict__ c, int n) {
  int row  = blockIdx.x * 8 + (threadIdx.x >> 5);
  int lane = threadIdx.x & 31;
  const float4* Wr = (const float4*)(W + (size_t)row * n);
  float s = 0.f;
  for (int i = lane; i < n / 4; i += 32) {
    float4 v = Wr[i];
    s += v.x + v.y + v.z + v.w;
  }
  for (int off = 16; off; off >>= 1) s += __shfl_down(s, off, 32);
  if (lane == 0) c[row] = s;
}

// ---- kernel 3: fused per-row dot (WMMA) + GELU + broadcast add -----------
// Each wave async-loads exactly the K-slice it consumes, waits only on its
// own ASYNCcnt, and runs its WMMA reduction immediately (no pre-compute
// barrier). Cross-wave visibility is established at the reduction barrier.
__global__ void __launch_bounds__(256)
fused_rows(const float* __restrict__ x, const float* __restrict__ ws,
           float* __restrict__ out) {
  __shared__ float smem[SMEM_FLOATS];
  const int tid  = threadIdx.x;
  const int lane = tid & 31;
  const int w    = tid >> 5;
  const long m0  = (long)blockIdx.x * ROWS;

  // Low 32 bits of the generic pointer == LDS byte offset (aperture mapping)
  const uint32_t lds_base = (uint32_t)(uintptr_t)(void*)smem;

  const int kbeg  = w * (K_DIM / 8);   // 512-float K slice owned by this wave
  const int f4beg = kbeg / 4;          // 128 float4s

  // Stage this wave's c slice (2KB) via async DMA
  const float* cg = ws + 64;
  for (int i = 0; i < 4; ++i) {
    int f4 = f4beg + lane + 32 * i;
    async_ld_b128(lds_base + (uint32_t)(C_BASE + f4 * 4) * 4u, cg + f4 * 4);
  }
  // Stage this wave's x slice: 8 rows x 512 floats (16KB) via async DMA
  for (int r = 0; r < ROWS; ++r) {
    const float* gr = x + (m0 + r) * (long)K_DIM;
    const uint32_t lr = lds_base + (uint32_t)(X_BASE + r * LDS_STRIDE) * 4u;
    for (int i = 0; i < 4; ++i) {
      int f4 = f4beg + lane + 32 * i;
      async_ld_b128(lr + (uint32_t)f4 * 16u, gr + f4 * 4);
    }
  }
  wait_async0();   // per-wave: this wave reads only LDS it wrote

  const float bt = ws[0];

  // WMMA dot over this wave's slice.
  // A(16x4 f32): lane<16 -> row=lane, k={k0,k0+1}; lane>=16 -> row=lane-16, k={k0+2,k0+3}
  //   rows 8..15 duplicate rows 0..7 (harmless: D rows are independent, never read)
  // B(4x16 f32): all 16 columns equal the c-chunk  =>  D[i][*] = dot partial of row i
  v8f acc = {};
  const int hi   = lane >> 4;
  const int mrow = lane & 15;
  const float* xs = smem + X_BASE + (mrow & (ROWS - 1)) * LDS_STRIDE;
  const float* cs = smem + C_BASE;
  for (int k0 = kbeg; k0 < kbeg + K_DIM / 8; k0 += 4) {
    const int kk = k0 + 2 * hi;
    v2f a; a.x = xs[kk]; a.y = xs[kk + 1];
    v2f b; b.x = cs[kk]; b.y = cs[kk + 1];
    acc = __builtin_amdgcn_wmma_f32_16x16x4_f32(false, a, false, b,
                                                (short)0, acc, false, false);
  }
  // lane 0: VGPR v holds D[M=v][N=0] == dot partial of row v (rows 0..7)
  if (lane == 0) {
    for (int v = 0; v < 8; ++v) smem[P_BASE + w * 8 + v] = acc[v];
  }
  __syncthreads();
  if (tid < ROWS) {
    float s = 0.f;
    for (int ww = 0; ww < 8; ++ww) s += smem[P_BASE + ww * 8 + tid];
    smem[R_BASE + tid] = gelu_tanh(s * (1.0f / (float)N_DIM) + bt);
  }
  __syncthreads();

  // out = x + row_scalar, streamed from LDS (x never re-read from HBM)
  for (int i = 0; i < ROWS * (K_DIM / 4) / 256; ++i) {
    int f4 = tid + 256 * i;
    int r  = f4 >> 10;
    int c4 = f4 & 1023;
    const float* src = smem + X_BASE + r * LDS_STRIDE + c4 * 4;
    const float rs = smem[R_BASE + r];
    float4 vv = { src[0] + rs, src[1] + rs, src[2] + rs, src[3] + rs };
    *(float4*)(out + (m0 + r) * (long)K_DIM + c4 * 4) = vv;
  }
}

extern "C" void kernel_launch(void* const* d_in, const int* in_sizes, int n_in,
                              void* d_out, int out_size, void* d_ws, size_t ws_size,
                              hipStream_t stream) {
  const float* x    = (const float*)d_in[0];
  const float* wgt  = (const float*)d_in[1];
  const float* bias = (const float*)d_in[2];
  const float* sub  = (const float*)d_in[3];
  float* out = (float*)d_out;
  float* ws  = (float*)d_ws;   // [0]: bias_term, [64..64+K): c

  const int  N = in_sizes[2];
  const int  K = in_sizes[1] / N;
  const long M = (long)in_sizes[0] / K;

  bias_term_kernel<<<1, 256, 0, stream>>>(bias, sub, ws, N);
  colsum_kernel<<<dim3(K / 8), 256, 0, stream>>>(wgt, ws + 64, N);
  fused_rows<<<dim3((unsigned)(M / ROWS)), 256, 0, stream>>>(x, ws, out);
}